// GraphDocSelfAttention_84464826843201
// MI455X (gfx1250) — compile-verified
//
#include <hip/hip_runtime.h>
#include <math.h>

#define Bsz 2
#define Lsz 1024
#define HIDsz 768
#define Hsz 12
#define TOPKsz 32
#define Dsz 64

typedef float v2f __attribute__((ext_vector_type(2)));
typedef float v8f __attribute__((ext_vector_type(8)));
typedef unsigned int v4u __attribute__((ext_vector_type(4)));
typedef int v8i __attribute__((ext_vector_type(8)));
typedef int v4i __attribute__((ext_vector_type(4)));

// ---------------------------------------------------------------------------
// TDM: async 2-D tile copy global -> LDS with hardware row padding.
// D# packing per CDNA5 ISA 8.3/8.4. tensor dims == tile dims (tile is always
// fully in-bounds for our shapes), tensor_dim0_stride = source leading dim.
// ---------------------------------------------------------------------------
__device__ __forceinline__ void tdm_load_2d(const void* gaddr, unsigned lds_addr,
                                            unsigned tile_d0, unsigned tile_d1,
                                            unsigned stride0_elems,
                                            unsigned pad_interval, unsigned pad_amount)
{
    const unsigned long long ga = (unsigned long long)gaddr;
    v4u g0;
    g0.x = 1u;                                            // count=1, user mode
    g0.y = lds_addr;                                      // LDS byte address
    g0.z = (unsigned)ga;                                  // global_addr[31:0]
    g0.w = (unsigned)((ga >> 32) & 0x01FFFFFFu)           // global_addr[56:32]
         | (2u << 30);                                    // type = 2 ("image")
    v8i g1;
    g1[0] = (int)((2u << 16)                              // data_size = 4 bytes
                | (1u << 20)                              // pad_enable
                | (pad_interval << 22)
                | (pad_amount << 25));
    g1[1] = (int)((tile_d0 & 0xFFFFu) << 16);             // tensor_dim0[15:0]
    g1[2] = (int)(((tile_d0 >> 16) & 0xFFFFu)             // tensor_dim0[31:16]
                | ((tile_d1 & 0xFFFFu) << 16));           // tensor_dim1[15:0]
    g1[3] = (int)(((tile_d1 >> 16) & 0xFFFFu)             // tensor_dim1[31:16]
                | ((tile_d0 & 0xFFFFu) << 16));           // tile_dim0
    g1[4] = (int)(tile_d1 & 0xFFFFu);                     // tile_dim1 (tile_dim2=0)
    g1[5] = (int)stride0_elems;                           // tensor_dim0_stride[31:0]
    g1[6] = 0;                                            // stride0[47:32] | stride1 lo
    g1[7] = 0;
    const v4i z4 = {0, 0, 0, 0};
#if __clang_major__ >= 23
    const v8i z8 = {0, 0, 0, 0, 0, 0, 0, 0};
    __builtin_amdgcn_tensor_load_to_lds(g0, g1, z4, z4, z8, 0);
#else
    __builtin_amdgcn_tensor_load_to_lds(g0, g1, z4, z4, 0);
#endif
}

// ---------------------------------------------------------------------------
// WMMA fp32 GEMM: C[M,N] = A[M,K] @ B[K,N] (+ optional bias[col]).
// Block tile 128x128, 256 threads = 8 wave32 waves, each wave owns 32x64
// (2x4 grid of 16x16 WMMA subtiles). K staged 16 at a time by the Tensor
// Data Mover into double-buffered LDS; wave 0 issues, TENSORcnt + barrier
// synchronize. M,N multiples of 128, K multiple of 16 (true for our shapes).
// ---------------------------------------------------------------------------
__global__ __launch_bounds__(256)
void gemm_f32_wmma(const float* __restrict__ A, const float* __restrict__ Bm,
                   float* __restrict__ C, const float* __restrict__ bias,
                   int M, int N, int K)
{
    __shared__ float As[2][128][20];   // TDM pads each 16-DWORD row by 4 DWORDs
    __shared__ float Bs[2][16][132];   // TDM pads each 128-DWORD row by 4 DWORDs

    const int t    = threadIdx.x;
    const int lane = t & 31;
    const int wave = t >> 5;
    const int wrow = (wave & 3) * 32;   // 0,32,64,96
    const int wcol = (wave >> 2) * 64;  // 0,64
    const int bm   = blockIdx.y * 128;
    const int bn   = blockIdx.x * 128;

    const unsigned asb0 = (unsigned)(uintptr_t)&As[0][0][0];
    const unsigned asb1 = (unsigned)(uintptr_t)&As[1][0][0];
    const unsigned bsb0 = (unsigned)(uintptr_t)&Bs[0][0][0];
    const unsigned bsb1 = (unsigned)(uintptr_t)&Bs[1][0][0];

    auto issue_tiles = [&](int it, int bf) {
        if (wave != 0) return;
        // A tile: 128 rows x 16 floats, row stride K; pad 16 DWORDs -> +4 (enc 3,3)
        tdm_load_2d(A + (size_t)bm * K + it * 16, bf ? asb1 : asb0,
                    16u, 128u, (unsigned)K, 3u, 3u);
        // B tile: 16 rows x 128 floats, row stride N; pad 128 DWORDs -> +4 (enc 6,3)
        tdm_load_2d(Bm + (size_t)it * 16 * N + bn, bf ? bsb1 : bsb0,
                    128u, 16u, (unsigned)N, 6u, 3u);
    };

    v8f acc[2][4];
    const v8f vzero = {0.f, 0.f, 0.f, 0.f, 0.f, 0.f, 0.f, 0.f};
#pragma unroll
    for (int sm = 0; sm < 2; ++sm)
#pragma unroll
        for (int sn = 0; sn < 4; ++sn) acc[sm][sn] = vzero;

    const int nk = K / 16;
    issue_tiles(0, 0);

    for (int it = 0; it < nk; ++it) {
        const int bf = it & 1;
        if (it + 1 < nk) {
            issue_tiles(it + 1, bf ^ 1);
            __builtin_amdgcn_s_wait_tensorcnt(2);  // tile `it` complete (in-order)
        } else {
            __builtin_amdgcn_s_wait_tensorcnt(0);
        }
        __syncthreads();   // LDS tile visible to all waves

#pragma unroll
        for (int kk = 0; kk < 16; kk += 4) {
            // A fragment: lanes 0-15 hold K={kk,kk+1}, lanes 16-31 K={kk+2,kk+3}
            const int kb = kk + ((lane >> 4) << 1);
            v2f afrag[2];
#pragma unroll
            for (int sm = 0; sm < 2; ++sm) {
                const int row = wrow + sm * 16 + (lane & 15);
                afrag[sm].x = As[bf][row][kb];
                afrag[sm].y = As[bf][row][kb + 1];
            }
            v2f bfrag[4];
#pragma unroll
            for (int sn = 0; sn < 4; ++sn) {
                const int col = wcol + sn * 16 + (lane & 15);
                bfrag[sn].x = Bs[bf][kb][col];
                bfrag[sn].y = Bs[bf][kb + 1][col];
            }
#pragma unroll
            for (int sm = 0; sm < 2; ++sm)
#pragma unroll
                for (int sn = 0; sn < 4; ++sn)
                    acc[sm][sn] = __builtin_amdgcn_wmma_f32_16x16x4_f32(
                        false, afrag[sm], false, bfrag[sn],
                        (short)0, acc[sm][sn], false, false);
        }
        __syncthreads();   // all waves done reading buf `bf` before TDM reuses it
    }

    // ---- epilogue: C/D layout -> VGPR r holds M=r (lanes 0-15) / M=r+8 ----
#pragma unroll
    for (int sm = 0; sm < 2; ++sm) {
#pragma unroll
        for (int sn = 0; sn < 4; ++sn) {
            const int col = bn + wcol + sn * 16 + (lane & 15);
            const float bv = bias ? bias[col] : 0.0f;
#pragma unroll
            for (int r = 0; r < 8; ++r) {
                const int row = bm + wrow + sm * 16 + r + ((lane >> 4) << 3);
                C[(size_t)row * N + col] = acc[sm][sn][r] + bv;
            }
        }
    }
}

// ---------------------------------------------------------------------------
// Q = qkv[:, 0:768] + q_bias ; V = qkv[:, 1536:2304] + v_bias
// ---------------------------------------------------------------------------
__global__ __launch_bounds__(256)
void build_qv(const float* __restrict__ qkv, const float* __restrict__ qb,
              const float* __restrict__ vb, float* __restrict__ Q,
              float* __restrict__ V)
{
    const int idx = blockIdx.x * 256 + threadIdx.x;     // < 2048*768
    const int row = idx / HIDsz;
    const int col = idx - row * HIDsz;
    const size_t base = (size_t)row * (3 * HIDsz);
    Q[idx] = qkv[base + col] + qb[col];
    V[idx] = qkv[base + 2 * HIDsz + col] + vb[col];
}

__device__ __forceinline__ float dot64(const float* __restrict__ a,
                                       const float* __restrict__ b)
{
    float s = 0.f;
#pragma unroll
    for (int i = 0; i < 16; ++i) {
        const float4 x = ((const float4*)a)[i];
        const float4 y = ((const float4*)b)[i];
        s += x.x * y.x + x.y * y.y + x.z * y.z + x.w * y.w;
    }
    return s;
}

// ---------------------------------------------------------------------------
// Sparse attention: one wave32 per (b,h,i) row, one lane per TOPK slot.
// Everything outside the TOPK index set is -1e8 in the reference and
// underflows to exactly 0 after softmax, so only these 32 slots matter.
// ---------------------------------------------------------------------------
__global__ __launch_bounds__(256)
void attn_sparse(const float* __restrict__ qkv_raw,  // (2048, 2304): k at col 768
                 const float* __restrict__ Q,        // (2048, 768) q+bias
                 const float* __restrict__ V,        // (2048, 768) v+bias
                 const float* __restrict__ QB,       // (2048, 768) q@w_rq+b_rq
                 const float* __restrict__ rbe,      // (B,L,TOPK,HID)
                 const int*   __restrict__ ridx,     // (B,L,TOPK)
                 const unsigned char* __restrict__ amask, // (B,1,L,L)
                 const float* __restrict__ rel,      // (B,H,L,L)
                 const float* __restrict__ rel2,     // (B,H,L,L)
                 float* __restrict__ out)            // (B,L,HID)
{
    const int r    = blockIdx.x * 8 + (threadIdx.x >> 5);  // < B*H*L
    const int lane = threadIdx.x & 31;
    const int b    = r / (Hsz * Lsz);
    const int rem  = r - b * Hsz * Lsz;
    const int h    = rem / Lsz;
    const int i    = rem - h * Lsz;
    const float scale = 0.125f;  // 1/sqrt(64)

    const int j = ridx[((size_t)(b * Lsz + i)) * TOPKsz + lane];

    // base QK score for this lane's key index
    const float* qrow = Q + (size_t)(b * Lsz + i) * HIDsz + h * Dsz;
    const float* krow = qkv_raw + (size_t)(b * Lsz + j) * (3 * HIDsz) + HIDsz + h * Dsz;
    const float base  = scale * dot64(qrow, krow);

    // bbox score for this lane's slot
    const float* qbrow = QB + (size_t)(b * Lsz + i) * HIDsz + h * Dsz;
    const float* rrow  = rbe + ((size_t)(b * Lsz + i) * TOPKsz + lane) * HIDsz + h * Dsz;
    const float bbox   = scale * dot64(qbrow, rrow);

    const size_t roff = ((size_t)(b * Hsz + h) * Lsz + i) * Lsz + j;
    const float relv  = rel[roff] + rel2[roff];
    const unsigned char mk = amask[((size_t)b * Lsz + i) * Lsz + j];

    // merge duplicate indices (reference scatter-add accumulates them)
    float bsum = 0.f;
    bool first = true;
#pragma unroll
    for (int u = 0; u < 32; ++u) {
        const int   ju = __shfl(j, u, 32);
        const float bu = __shfl(bbox, u, 32);
        if (ju == j) {
            bsum += bu;
            if (u < lane) first = false;
        }
    }

    float s;
    if (first) {
        s = base + bsum + relv;
        if (mk) s = -100000000.0f;   // attention_mask applied last
    } else {
        s = -INFINITY;               // duplicate slot: excluded
    }

    // wave softmax over active lanes
    float m = s;
#pragma unroll
    for (int o = 16; o; o >>= 1) m = fmaxf(m, __shfl_xor(m, o, 32));
    float p = first ? __expf(s - m) : 0.f;
    float sum = p;
#pragma unroll
    for (int o = 16; o; o >>= 1) sum += __shfl_xor(sum, o, 32);
    p /= sum;

    // ctx[i, :] = sum_j p_j * v[j, :]  (lane covers dims lane and lane+32)
    float a0 = 0.f, a1 = 0.f;
    for (int u = 0; u < 32; ++u) {
        const float pu = __shfl(p, u, 32);
        const int   ju = __shfl(j, u, 32);
        if (pu > 0.f) {   // wave-uniform: skips duplicate / underflowed slots
            const float* vrow = V + (size_t)(b * Lsz + ju) * HIDsz + h * Dsz;
            a0 += pu * vrow[lane];
            a1 += pu * vrow[lane + 32];
        }
    }
    float* orow = out + (size_t)(b * Lsz + i) * HIDsz + h * Dsz;
    orow[lane]      = a0;
    orow[lane + 32] = a1;
}

extern "C" void kernel_launch(void* const* d_in, const int* in_sizes, int n_in,
                              void* d_out, int out_size, void* d_ws, size_t ws_size,
                              hipStream_t stream) {
    (void)in_sizes; (void)n_in; (void)out_size; (void)ws_size;
    const float* hs    = (const float*)d_in[0];
    const float* rbe   = (const float*)d_in[1];
    const int*   ridx  = (const int*)d_in[2];
    const unsigned char* amask = (const unsigned char*)d_in[3];
    const float* rel   = (const float*)d_in[4];
    const float* rel2  = (const float*)d_in[5];
    const float* wqkv  = (const float*)d_in[6];
    const float* qbias = (const float*)d_in[7];
    const float* vbias = (const float*)d_in[8];
    const float* wrq   = (const float*)d_in[9];
    const float* brq   = (const float*)d_in[10];
    float* out = (float*)d_out;

    const int M = Bsz * Lsz;          // 2048
    float* ws      = (float*)d_ws;
    float* qkv_raw = ws;                                  // 2048 x 2304
    float* Q       = qkv_raw + (size_t)M * 3 * HIDsz;     // 2048 x 768
    float* V       = Q + (size_t)M * HIDsz;               // 2048 x 768
    float* QB      = V + (size_t)M * HIDsz;               // 2048 x 768

    // 1) qkv = hs @ w_qkv
    gemm_f32_wmma<<<dim3((3 * HIDsz) / 128, M / 128), 256, 0, stream>>>(
        hs, wqkv, qkv_raw, nullptr, M, 3 * HIDsz, HIDsz);
    // 2) Q = q + q_bias ; V = v + v_bias
    build_qv<<<(M * HIDsz) / 256, 256, 0, stream>>>(qkv_raw, qbias, vbias, Q, V);
    // 3) QB = Q @ w_rq + b_rq
    gemm_f32_wmma<<<dim3(HIDsz / 128, M / 128), 256, 0, stream>>>(
        Q, wrq, QB, brq, M, HIDsz, HIDsz);
    // 4) sparse attention + output
    attn_sparse<<<(Bsz * Hsz * Lsz) / 8, 256, 0, stream>>>(
        qkv_raw, Q, V, QB, rbe, ridx, amask, rel, rel2, out);
}